// a_max_op_52793738003170
// MI455X (gfx1250) — compile-verified
//
#include <hip/hip_runtime.h>
#include <float.h>

typedef __attribute__((ext_vector_type(2))) float v2f;
typedef __attribute__((ext_vector_type(8))) float v8f;

#define GNN_D 128

// Native CDNA5 float max atomic, no-return form (STOREcnt-tracked, no CAS loop).
__device__ __forceinline__ void atomic_max_f32(float* addr, float val) {
  asm volatile("global_atomic_max_num_f32 %0, %1, off"
               :
               : "v"(addr), "v"(val)
               : "memory");
}

// ---------------- init: agg = -FLT_MAX, indeg = 0 ----------------
__global__ void gnn_init_kernel(float* __restrict__ agg, int* __restrict__ indeg,
                                int n_nodes) {
  const size_t total = (size_t)n_nodes * GNN_D;
  const size_t stride = (size_t)gridDim.x * blockDim.x;
  for (size_t i = (size_t)blockIdx.x * blockDim.x + threadIdx.x; i < total; i += stride)
    agg[i] = -FLT_MAX;
  for (size_t j = (size_t)blockIdx.x * blockDim.x + threadIdx.x; j < (size_t)n_nodes; j += stride)
    indeg[j] = 0;
}

// ---------------- GEMM: h2 = relu(h @ W^T + b), WMMA f32 16x16x4 ----------------
// One wave computes a 16-row x 128-col slab: 8 accumulators (16x16 tiles),
// A fragment loaded once per k-step and reused for all 8 WMMAs.
// Layout (ISA 7.12.2, 32-bit): lane m = lane&15, group g = lane>>4,
//   A: VGPR v holds K = k0 + v + 2*g for row m      -> float2 load at h[row, k0+2g]
//   B: VGPR v holds K = k0 + v + 2*g for col n      -> B[k][n] = W[n][k], float2 at W[col, k0+2g]
//   C/D: VGPR i, group g -> row = i + 8*g, col = lane&15
__global__ __launch_bounds__(256) void gnn_gemm_relu_kernel(
    const float* __restrict__ h, const float* __restrict__ W,
    const float* __restrict__ bias, float* __restrict__ h2, int n_mtiles) {
  const int wave = (int)(((size_t)blockIdx.x * blockDim.x + threadIdx.x) >> 5);
  const int lane = threadIdx.x & 31;
  if (wave >= n_mtiles) return;

  const int m = lane & 15;
  const int g = lane >> 4;
  const int row0 = wave * 16;

  v8f acc[8] = {};

  const float* arow = h + (size_t)(row0 + m) * GNN_D;

  for (int k0 = 0; k0 < GNN_D; k0 += 4) {
    const v2f a = *(const v2f*)(arow + k0 + 2 * g);
#pragma unroll
    for (int n = 0; n < 8; ++n) {
      // B fragment: col = n*16 + m, elements W[col, k0+2g .. k0+2g+1]
      const v2f bfrag = *(const v2f*)(W + (size_t)(n * 16 + m) * GNN_D + k0 + 2 * g);
      acc[n] = __builtin_amdgcn_wmma_f32_16x16x4_f32(
          /*neg_a=*/false, a, /*neg_b=*/false, bfrag,
          /*c_mod=*/(short)0, acc[n], /*reuse_a=*/false, /*reuse_b=*/false);
    }
  }

  // epilogue: bias + relu + store
#pragma unroll
  for (int n = 0; n < 8; ++n) {
    const int col = n * 16 + m;
    const float bb = bias[col];
#pragma unroll
    for (int i = 0; i < 8; ++i) {
      const int row = row0 + i + 8 * g;
      float v = acc[n][i] + bb;
      h2[(size_t)row * GNN_D + col] = v > 0.f ? v : 0.f;
    }
  }
}

// ---------------- edge phase: gather h2[src], atomic max into agg[dst] ----------------
// One wave per edge; lane covers 4 features (32 lanes * 4 = 128). h2 and agg are
// L2-resident (25.6 MB each, 192 MB L2) so the random traffic never hits HBM.
__global__ __launch_bounds__(256) void gnn_edge_kernel(
    const float* __restrict__ h2, const int* __restrict__ src,
    const int* __restrict__ dst, float* __restrict__ agg,
    int* __restrict__ indeg, int n_edges) {
  const int wave = (int)(((size_t)blockIdx.x * blockDim.x + threadIdx.x) >> 5);
  const int lane = threadIdx.x & 31;
  if (wave >= n_edges) return;

  const int s = src[wave];
  const int t = dst[wave];
  if (lane == 0) atomicAdd(indeg + t, 1);

  const float4 v = *(const float4*)(h2 + (size_t)s * GNN_D + lane * 4);
  float* ap = agg + (size_t)t * GNN_D + lane * 4;
  atomic_max_f32(ap + 0, v.x);
  atomic_max_f32(ap + 1, v.y);
  atomic_max_f32(ap + 2, v.z);
  atomic_max_f32(ap + 3, v.w);
}

// ---------------- finalize: out = indeg>0 ? agg : h2 (h2 already in out) ----------------
__global__ void gnn_finalize_kernel(const float* __restrict__ agg,
                                    const int* __restrict__ indeg,
                                    float* __restrict__ out, int n_nodes) {
  const size_t tid = (size_t)blockIdx.x * blockDim.x + threadIdx.x;
  const size_t total = (size_t)n_nodes * (GNN_D / 4);
  if (tid >= total) return;
  const int node = (int)(tid >> 5);  // 32 float4 chunks per node
  if (indeg[node] > 0) {
    ((float4*)out)[tid] = ((const float4*)agg)[tid];
  }
}

extern "C" void kernel_launch(void* const* d_in, const int* in_sizes, int n_in,
                              void* d_out, int out_size, void* d_ws, size_t ws_size,
                              hipStream_t stream) {
  const float* h   = (const float*)d_in[0];
  const float* W   = (const float*)d_in[1];
  const float* b   = (const float*)d_in[2];
  const int*   src = (const int*)d_in[3];
  const int*   dst = (const int*)d_in[4];
  float* out = (float*)d_out;

  const int n_nodes = in_sizes[0] / GNN_D;  // 50000
  const int n_edges = in_sizes[3];          // 800000

  // workspace layout: agg[n_nodes*128] f32, then indeg[n_nodes] i32
  float* agg  = (float*)d_ws;
  int* indeg  = (int*)((char*)d_ws + (size_t)n_nodes * GNN_D * sizeof(float));

  {  // init aggregate buffer + in-degree counters
    const int threads = 256;
    const int blocks = (int)(((size_t)n_nodes * GNN_D + threads - 1) / threads);
    gnn_init_kernel<<<blocks, threads, 0, stream>>>(agg, indeg, n_nodes);
  }
  {  // h2 = relu(h @ W^T + b) into d_out via WMMA
    const int n_mtiles = n_nodes / 16;  // 3125 (50000 divisible by 16)
    const int waves_per_block = 8;      // 256 threads, wave32
    const int blocks = (n_mtiles + waves_per_block - 1) / waves_per_block;
    gnn_gemm_relu_kernel<<<blocks, 256, 0, stream>>>(h, W, b, out, n_mtiles);
  }
  {  // per-edge gather + native float-max atomics + in-degree count
    const int waves_per_block = 8;
    const int blocks = (n_edges + waves_per_block - 1) / waves_per_block;
    gnn_edge_kernel<<<blocks, 256, 0, stream>>>(out, src, dst, agg, indeg, n_edges);
  }
  {  // conditional overwrite with aggregate
    const int threads = 256;
    const size_t total = (size_t)n_nodes * (GNN_D / 4);
    const int blocks = (int)((total + threads - 1) / threads);
    gnn_finalize_kernel<<<blocks, threads, 0, stream>>>(agg, indeg, out, n_nodes);
  }
}